// DGCNN_51788715655869
// MI455X (gfx1250) — compile-verified
//
#include <hip/hip_runtime.h>

typedef __attribute__((ext_vector_type(2))) float v2f;
typedef __attribute__((ext_vector_type(4))) float v4f;
typedef __attribute__((ext_vector_type(8))) float v8f;

#define GR  4096
#define MN  50     // nodes per graph
#define KP  30     // sort-pool k
#define HC  32
#define FIN 128
#define EPG 400    // edges per graph (M*DEG)
#define DLAT 97
#define APAD 132   // x-tile row stride (bank-conflict-free, 16B aligned)
#define HPAD 36    // H row stride

__launch_bounds__(256)
__global__ void dgcnn_fused_kernel(const float* __restrict__ x,
                                   const int*   __restrict__ ei,
                                   const float* __restrict__ W0, const float* __restrict__ b0,
                                   const float* __restrict__ W1, const float* __restrict__ b1,
                                   const float* __restrict__ W2, const float* __restrict__ b2,
                                   const float* __restrict__ W3, const float* __restrict__ b3,
                                   const float* __restrict__ C1w, const float* __restrict__ C1b,
                                   const float* __restrict__ C2w, const float* __restrict__ C2b,
                                   const float* __restrict__ L1w, const float* __restrict__ L1b,
                                   const float* __restrict__ L2w, const float* __restrict__ L2b,
                                   float* __restrict__ out, int E_total)
{
    __shared__ float Ax[64][APAD];    // staged x tile, rows 50..63 zero
    __shared__ float HW[64][HPAD];    // matmul output tile
    __shared__ float Hin[64][HPAD];   // layer input (rows 50..63 zero)
    __shared__ float lat[50][97];     // concatenated latents
    __shared__ int   eLs[EPG];
    __shared__ int   eLd[EPG];
    __shared__ int   esrc[EPG];       // CSR-ordered source ids
    __shared__ float enorm[EPG];      // CSR-ordered edge norms
    __shared__ int   cnt[50];
    __shared__ int   base[51];
    __shared__ float dinv[50];
    __shared__ float HW4[64];
    __shared__ int   sel[KP];
    __shared__ float z1[16 * 30];
    __shared__ float pbuf[16 * 15];
    __shared__ float flat[352];
    __shared__ float red[128];

    const int tid  = threadIdx.x;
    const int g    = blockIdx.x;
    const int lane = tid & 31;
    const int wave = tid >> 5;

    // ---------- init: degree counts, zero-pad rows of Hin ----------
    if (tid < 50) cnt[tid] = 0;
    for (int t = tid; t < 14 * 32; t += 256) Hin[50 + (t >> 5)][t & 31] = 0.0f;

    // ---------- stage x tile into LDS (coalesced float4), zero rows 50..63 ----------
    for (int j = tid; j < 64 * 32; j += 256) {
        int row = j >> 5, c4 = j & 31;
        v4f v = {0.0f, 0.0f, 0.0f, 0.0f};
        if (row < MN)
            v = *(const v4f*)(x + (size_t)(g * MN + row) * FIN + c4 * 4);
        *(v4f*)(&Ax[row][c4 * 4]) = v;
    }

    // warm L2 for the dense head weights (global_prefetch_b8)
    for (int off = tid * 32; off < 352 * 128; off += 256 * 32)
        __builtin_prefetch(L1w + off, 0, 1);

    __syncthreads();

    // ---------- load edges (graph-local), count in-degree ----------
    for (int e = tid; e < EPG; e += 256) {
        int s = ei[g * EPG + e]           - g * MN;
        int d = ei[E_total + g * EPG + e] - g * MN;
        eLs[e] = s;
        eLd[e] = d;
        atomicAdd(&cnt[d], 1);
    }
    __syncthreads();

    if (tid < 50) dinv[tid] = rsqrtf((float)(cnt[tid] + 1));   // +1 self loop
    if (tid <= 50) {
        int s = 0;
        for (int j = 0; j < tid && j < 50; ++j) s += cnt[j];
        base[tid] = s;
    }
    __syncthreads();

    // ---------- stable CSR build: one thread per dst scans edges in order ----------
    if (tid < 50) {
        int idx = base[tid];
        float dv = dinv[tid];
        for (int e = 0; e < EPG; ++e) {
            if (eLd[e] == tid) {
                int s = eLs[e];
                esrc[idx]  = s;
                enorm[idx] = dinv[s] * dv;
                ++idx;
            }
        }
    }
    __syncthreads();

    // ---------- WMMA tile assignment (wave -> 16x16 output tile) ----------
    const int mtile = wave >> 1;           // 0..3  (M = 64 rows)
    const int ntile = wave & 1;            // 0..1  (N = 32 cols)
    const int r16   = lane & 15;
    const int kh    = (lane >> 4) << 1;    // 0 or 2 : K sub-offset per ISA A/B layout
    const int mrow  = mtile * 16 + r16;
    const int ncol  = ntile * 16 + r16;
    const int rowhi = (lane >> 4) << 3;    // C/D: VGPR r -> rows r, r+8

    // ---------- layer 0: x(50x128) @ W0(128x32), fp32 WMMA 16x16x4 ----------
    {
        v8f acc = {};
        for (int kb = 0; kb < FIN / 4; ++kb) {
            int k0 = kb * 4 + kh;
            v2f a, b;
            a.x = Ax[mrow][k0];
            a.y = Ax[mrow][k0 + 1];
            b.x = W0[k0 * HC + ncol];
            b.y = W0[(k0 + 1) * HC + ncol];
            acc = __builtin_amdgcn_wmma_f32_16x16x4_f32(false, a, false, b,
                                                        (short)0, acc, false, false);
        }
        for (int r = 0; r < 8; ++r)
            HW[mtile * 16 + r + rowhi][ntile * 16 + r16] = acc[r];
    }
    __syncthreads();

    // aggregate + bias + tanh -> Hin, lat[:,0:32]
    for (int t = tid; t < MN * HC; t += 256) {
        int d = t >> 5, ch = t & 31;
        float a = HW[d][ch] * dinv[d] * dinv[d];          // self loop
        int e1 = base[d + 1];
        for (int p = base[d]; p < e1; ++p) a += HW[esrc[p]][ch] * enorm[p];
        float v = tanhf(a + b0[ch]);
        Hin[d][ch] = v;
        lat[d][ch] = v;
    }
    __syncthreads();

    // ---------- layers 1,2: Hin(64x32) @ W(32x32) ----------
    for (int L = 0; L < 2; ++L) {
        const float* Wl = (L == 0) ? W1 : W2;
        const float* bl = (L == 0) ? b1 : b2;
        v8f acc = {};
        for (int kb = 0; kb < HC / 4; ++kb) {
            int k0 = kb * 4 + kh;
            v2f a, b;
            a.x = Hin[mrow][k0];
            a.y = Hin[mrow][k0 + 1];
            b.x = Wl[k0 * HC + ncol];
            b.y = Wl[(k0 + 1) * HC + ncol];
            acc = __builtin_amdgcn_wmma_f32_16x16x4_f32(false, a, false, b,
                                                        (short)0, acc, false, false);
        }
        for (int r = 0; r < 8; ++r)
            HW[mtile * 16 + r + rowhi][ntile * 16 + r16] = acc[r];
        __syncthreads();

        for (int t = tid; t < MN * HC; t += 256) {
            int d = t >> 5, ch = t & 31;
            float a = HW[d][ch] * dinv[d] * dinv[d];
            int e1 = base[d + 1];
            for (int p = base[d]; p < e1; ++p) a += HW[esrc[p]][ch] * enorm[p];
            float v = tanhf(a + bl[ch]);
            Hin[d][ch] = v;
            lat[d][(L + 1) * HC + ch] = v;
        }
        __syncthreads();
    }

    // ---------- layer 3: Hin(50x32) @ W3(32x1) ----------
    if (tid < MN) {
        float s = 0.0f;
        for (int c = 0; c < HC; ++c) s += Hin[tid][c] * W3[c];
        HW4[tid] = s;
    }
    __syncthreads();
    if (tid < MN) {
        float a = HW4[tid] * dinv[tid] * dinv[tid];
        int e1 = base[tid + 1];
        for (int p = base[tid]; p < e1; ++p) a += HW4[esrc[p]] * enorm[p];
        lat[tid][96] = tanhf(a + b3[0]);
    }
    __syncthreads();

    // ---------- sort-pool: stable rank by last channel, descending ----------
    if (tid < MN) {
        float v = lat[tid][96];
        int rank = 0;
        for (int j = 0; j < MN; ++j) {
            float vj = lat[j][96];
            rank += (vj > v) || (vj == v && j < tid);
        }
        if (rank < KP) sel[rank] = tid;
    }
    __syncthreads();

    // ---------- conv1d(1,16,97,stride=97) + relu ----------
    for (int i = tid; i < 16 * KP; i += 256) {
        int c = i / KP, t = i % KP;
        int node = sel[t];
        float s = C1b[c];
        for (int d = 0; d < DLAT; ++d) s += lat[node][d] * C1w[c * DLAT + d];
        z1[c * KP + t] = fmaxf(s, 0.0f);
    }
    __syncthreads();

    // ---------- maxpool1d(2,2) ----------
    for (int i = tid; i < 16 * 15; i += 256) {
        int c = i / 15, t = i % 15;
        pbuf[i] = fmaxf(z1[c * KP + 2 * t], z1[c * KP + 2 * t + 1]);
    }
    __syncthreads();

    // ---------- conv1d(16,32,5) + relu, flatten [32,11] ----------
    for (int i = tid; i < 32 * 11; i += 256) {
        int o = i / 11, t = i % 11;
        float s = C2b[o];
        for (int ii = 0; ii < 16; ++ii)
            for (int w = 0; w < 5; ++w)
                s += pbuf[ii * 15 + t + w] * C2w[o * 80 + ii * 5 + w];
        flat[i] = fmaxf(s, 0.0f);
    }
    __syncthreads();

    // ---------- dense 352->128 relu, dense 128->1 ----------
    if (tid < 128) {
        float s = L1b[tid];
        for (int f = 0; f < 352; ++f) s += flat[f] * L1w[f * 128 + tid];
        red[tid] = fmaxf(s, 0.0f) * L2w[tid];
    }
    __syncthreads();
    if (tid == 0) {
        float s = L2b[0];
        for (int j = 0; j < 128; ++j) s += red[j];
        out[g] = s;
    }
}

extern "C" void kernel_launch(void* const* d_in, const int* in_sizes, int n_in,
                              void* d_out, int out_size, void* d_ws, size_t ws_size,
                              hipStream_t stream) {
    const float* x   = (const float*)d_in[0];
    const int*   ei  = (const int*)  d_in[1];
    // d_in[2] = batch (unused: graphs are contiguous equal-size blocks)
    const float* W0  = (const float*)d_in[3];
    const float* b0  = (const float*)d_in[4];
    const float* W1  = (const float*)d_in[5];
    const float* b1  = (const float*)d_in[6];
    const float* W2  = (const float*)d_in[7];
    const float* b2  = (const float*)d_in[8];
    const float* W3  = (const float*)d_in[9];
    const float* b3  = (const float*)d_in[10];
    const float* C1w = (const float*)d_in[11];
    const float* C1b = (const float*)d_in[12];
    const float* C2w = (const float*)d_in[13];
    const float* C2b = (const float*)d_in[14];
    const float* L1w = (const float*)d_in[15];
    const float* L1b = (const float*)d_in[16];
    const float* L2w = (const float*)d_in[17];
    const float* L2b = (const float*)d_in[18];

    int E_total = in_sizes[1] / 2;   // edge_index is [2, E]

    dgcnn_fused_kernel<<<GR, 256, 0, stream>>>(x, ei, W0, b0, W1, b1, W2, b2,
                                               W3, b3, C1w, C1b, C2w, C2b,
                                               L1w, L1b, L2w, L2b,
                                               (float*)d_out, E_total);
}